// Eikonal3D_22986664968779
// MI455X (gfx1250) — compile-verified
//
#include <hip/hip_runtime.h>

#define GN   128
#define GBIG 1.0e5f
#define NITER 128

// One Jacobi sweep of the upwind eikonal update.
// Block = (128 k, 4 j). Grid = (1, 32 j-tiles, 128 i-planes).
// Center-plane rows (j-1 .. j+4) are staged into LDS via the CDNA5 async
// global->LDS path (ASYNCcnt); i+-1 plane and f loads are issued before the
// async wait so their latency overlaps the staging + barrier.
__global__ __launch_bounds__(512) void eik_step_kernel(
    const float* __restrict__ usrc,
    const float* __restrict__ f,
    float* __restrict__ udst) {
  // 6 rows of 128 floats: rows map to j = jb-1 .. jb+4. Single shared alloc
  // => LDS offset 0 (confirmed in disasm), so async-to-LDS byte offsets are
  // computed directly.
  __shared__ float tile[6 * GN];

  const int tx = threadIdx.x;        // k index
  const int ty = threadIdx.y;        // j index within block (0..3)
  const int jb = blockIdx.y * 4;     // first j of this block
  const int i  = blockIdx.z;         // i-plane

  // ---- stage 6 rows (768 floats) into LDS with async global->LDS copies ----
  const int t = ty * GN + tx;        // 0..511
#pragma unroll
  for (int ee = 0; ee < 2; ++ee) {
    const int e = t + ee * 512;      // element 0..1023; valid when < 768
    if (e < 6 * GN) {
      const int r   = e >> 7;        // tile row 0..5
      const int col = e & (GN - 1);  // k
      const int jj  = jb + r - 1;    // global j (may be -1 or 128)
      if (jj >= 0 && jj < GN) {
        const unsigned goff = (((unsigned)i * GN + (unsigned)jj) * GN +
                               (unsigned)col) * 4u;   // byte offset into usrc
        const unsigned loff = (unsigned)e * 4u;       // LDS byte offset
        asm volatile("global_load_async_to_lds_b32 %0, %1, %2"
                     :: "v"(loff), "v"(goff), "s"(usrc)
                     : "memory");
      } else {
        tile[e] = GBIG;              // padded boundary neighbor
      }
    }
  }

  // ---- issue independent global loads NOW so they overlap the async wait ----
  const int j = jb + ty;
  const unsigned idx = ((unsigned)i * GN + (unsigned)j) * GN + (unsigned)tx;
  float im = GBIG, ip = GBIG;
  if (i > 0)      im = usrc[idx - GN * GN];
  if (i < GN - 1) ip = usrc[idx + GN * GN];
  const float fh = f[idx];   // H == 1.0

  // Wait for this wave's async LDS writes, then workgroup barrier
  // (barrier also covers the plain DS stores of BIG via DScnt).
  asm volatile("s_wait_asynccnt 0x0" ::: "memory");
  __syncthreads();

  // ---- per-cell upwind local solve ----
  const int row = ty + 1;
  const float c  = tile[row * GN + tx];
  const float km = (tx == 0)      ? GBIG : tile[row * GN + tx - 1];
  const float kp = (tx == GN - 1) ? GBIG : tile[row * GN + tx + 1];
  const float jm = tile[(row - 1) * GN + tx];
  const float jp = tile[(row + 1) * GN + tx];

  const float ax = fminf(im, ip);
  const float ay = fminf(jm, jp);
  const float az = fminf(km, kp);

  // median-of-3 sort network: a1 <= a2 <= a3
  const float t1 = fminf(ax, ay);
  const float t2 = fmaxf(ax, ay);
  const float a1 = fminf(t1, az);
  const float a3 = fmaxf(t2, az);
  const float a2 = fmaxf(t1, fminf(t2, az));

  // 1D update
  const float x1 = a1 + fh;
  // 2D update
  const float d12   = a1 - a2;
  const float disc2 = 2.0f * fh * fh - d12 * d12;
  const float x2    = 0.5f * (a1 + a2 + sqrtf(fmaxf(disc2, 0.0f)));
  // 3D update
  const float s     = a1 + a2 + a3;
  const float q     = a1 * a1 + a2 * a2 + a3 * a3;
  const float disc3 = s * s - 3.0f * (q - fh * fh);
  const float x3    = (s + sqrtf(fmaxf(disc3, 0.0f))) / 3.0f;

  const float x = (x1 <= a2) ? x1 : ((x2 <= a3) ? x2 : x3);

  udst[idx] = fminf(c, x);
}

extern "C" void kernel_launch(void* const* d_in, const int* in_sizes, int n_in,
                              void* d_out, int out_size, void* d_ws, size_t ws_size,
                              hipStream_t stream) {
  (void)in_sizes; (void)n_in; (void)out_size; (void)ws_size;

  const float* u0 = (const float*)d_in[0];
  const float* f  = (const float*)d_in[1];
  float* out = (float*)d_out;
  float* ws  = (float*)d_ws;   // ping buffer (>= 128^3 floats)

  const dim3 block(GN, 4, 1);
  const dim3 grid(1, GN / 4, GN);

  // Ping-pong: even sweeps write ws, odd sweeps write out.
  // Sweep 127 (odd) leaves the final result in d_out.
  const float* src = u0;
  for (int it = 0; it < NITER; ++it) {
    float* dst = (it & 1) ? out : ws;
    eik_step_kernel<<<grid, block, 0, stream>>>(src, f, dst);
    src = dst;
  }
}